// ErdosLoss_vertex_37374805410257
// MI455X (gfx1250) — compile-verified
//
#include <hip/hip_runtime.h>
#include <stdint.h>

// ---------------------------------------------------------------------------
// ErdosLoss: three scalar reductions over a big edge list.
//   expected_distance = sum_{e: r!=c} (1-p[r])(1-p[c]) / G
//   expected_weight   = sum(probs) / G          (segment_sum + mean collapses)
//   G                 = max(batch) + 1
// Memory-bound: 512MB edge stream dominates (roofline ~23us @ 23.3TB/s).
// probs (8MB) gathers are L2-resident *if* we don't let the index stream
// thrash L2 -> stream arrays use non-temporal loads (TH=NT), gathers use RT.
// Wave-level tail reduction uses V_WMMA_F32_16X16X4_F32 (reduce-by-ones).
// ---------------------------------------------------------------------------

typedef float v2f __attribute__((ext_vector_type(2)));
typedef float v8f __attribute__((ext_vector_type(8)));
typedef long long v2ll __attribute__((ext_vector_type(2)));

#if defined(__has_builtin)
#if __has_builtin(__builtin_amdgcn_wmma_f32_16x16x4_f32)
#define HAVE_WMMA_F32_16X16X4 1
#endif
#endif

// Sum of the 64 fp32 partials held by one wave32 (2 per lane).
// A (16x4 f32) layout: lanes 0-15 hold rows M=0..15 with VGPR0=K0,VGPR1=K1;
// lanes 16-31 hold K2,K3. B = all-ones 4x16 -> D[m][n] = rowsum(m).
// Lane n<16 sees M=0..7 in its 8 D VGPRs, lane n>=16 sees M=8..15,
// so full sum = (sum of my 8 D regs) + shfl_xor(.,16).
__device__ __forceinline__ float wave_sum2(float a0, float a1) {
#ifdef HAVE_WMMA_F32_16X16X4
  v2f A; A.x = a0; A.y = a1;
  v2f B; B.x = 1.0f; B.y = 1.0f;
  v8f C = {};
  v8f D = __builtin_amdgcn_wmma_f32_16x16x4_f32(
      /*neg_a=*/false, A, /*neg_b=*/false, B,
      /*c_mod=*/(short)0, C, /*reuse_a=*/false, /*reuse_b=*/false);
  float s = D[0] + D[1] + D[2] + D[3] + D[4] + D[5] + D[6] + D[7];
  s += __shfl_xor(s, 16, 32);
  return s;
#else
  float s = a0 + a1;
  for (int o = 16; o > 0; o >>= 1) s += __shfl_xor(s, o, 32);
  return s;
#endif
}

__device__ __forceinline__ void block_accum(float a0, float a1, double* gacc) {
  __shared__ float lds[8];  // 256 threads -> 8 wave32
  const int wave = threadIdx.x >> 5;
  const int lane = threadIdx.x & 31;
  float w = wave_sum2(a0, a1);  // EXEC is full here: whole block participates
  if (lane == 0) lds[wave] = w;
  __syncthreads();
  if (threadIdx.x == 0) {
    float t = 0.0f;
    const int nw = blockDim.x >> 5;
    for (int i = 0; i < nw; ++i) t += lds[i];
    atomicAdd(gacc, (double)t);
  }
}

__global__ void erdos_init_kernel(double* acc, int* bmax) {
  if (threadIdx.x == 0 && blockIdx.x == 0) {
    acc[0] = 0.0;  // edge sum
    acc[1] = 0.0;  // probs sum
    bmax[0] = 0;   // batch max (values are >= 0)
  }
}

__global__ void __launch_bounds__(256)
erdos_edge_kernel(const float* __restrict__ probs,
                  const long long* __restrict__ rows,
                  const long long* __restrict__ cols,
                  long long E, double* __restrict__ acc) {
  const long long tid = (long long)blockIdx.x * blockDim.x + threadIdx.x;
  const long long stride = (long long)gridDim.x * blockDim.x;
  float a0 = 0.0f, a1 = 0.0f;

  const bool aligned16 =
      (((uintptr_t)rows | (uintptr_t)cols) & 15u) == 0;

  if (aligned16) {
    // 2 edges per iteration; non-temporal b128 streaming loads of the
    // index arrays so they bypass-evict and leave L2 to the probs gathers.
    const v2ll* __restrict__ r2 = reinterpret_cast<const v2ll*>(rows);
    const v2ll* __restrict__ c2 = reinterpret_cast<const v2ll*>(cols);
    const long long nPairs = E >> 1;
    for (long long p = tid; p < nPairs; p += stride) {
      v2ll r = __builtin_nontemporal_load(&r2[p]);
      v2ll c = __builtin_nontemporal_load(&c2[p]);
      float pu0 = probs[r.x], pv0 = probs[c.x];   // RT: L2-resident gathers
      float pu1 = probs[r.y], pv1 = probs[c.y];
      float t0 = (1.0f - pu0) * (1.0f - pv0);
      float t1 = (1.0f - pu1) * (1.0f - pv1);
      a0 += (r.x != c.x) ? t0 : 0.0f;             // branchless self-loop mask
      a1 += (r.y != c.y) ? t1 : 0.0f;
    }
    if ((E & 1) && tid == 0) {
      long long ri = rows[E - 1], ci = cols[E - 1];
      if (ri != ci) a0 += (1.0f - probs[ri]) * (1.0f - probs[ci]);
    }
  } else {
    for (long long i = tid; i < E; i += stride) {
      long long ri = __builtin_nontemporal_load(&rows[i]);
      long long ci = __builtin_nontemporal_load(&cols[i]);
      float t = (1.0f - probs[ri]) * (1.0f - probs[ci]);
      a0 += (ri != ci) ? t : 0.0f;
    }
  }
  block_accum(a0, a1, acc);
}

__global__ void __launch_bounds__(256)
erdos_node_kernel(const float* __restrict__ probs,
                  const long long* __restrict__ batch,
                  long long N, double* __restrict__ accp,
                  int* __restrict__ bmax) {
  const long long tid = (long long)blockIdx.x * blockDim.x + threadIdx.x;
  const long long stride = (long long)gridDim.x * blockDim.x;

  // --- sum(probs): non-temporal b64 streaming loads (linear pass; the
  // edge kernel's gathers are what we want cached, not this sweep).
  float a0 = 0.0f, a1 = 0.0f;
  const v2f* __restrict__ p2 = reinterpret_cast<const v2f*>(probs);
  const long long nPairs = N >> 1;
  for (long long p = tid; p < nPairs; p += stride) {
    v2f v = __builtin_nontemporal_load(&p2[p]);
    a0 += v.x;
    a1 += v.y;
  }
  if ((N & 1) && tid == 0) a0 += probs[N - 1];

  // --- max(batch): non-temporal b128 streaming loads
  long long m = 0;
  const v2ll* __restrict__ b2 = reinterpret_cast<const v2ll*>(batch);
  for (long long p = tid; p < nPairs; p += stride) {
    v2ll b = __builtin_nontemporal_load(&b2[p]);
    long long mm = b.x > b.y ? b.x : b.y;
    m = mm > m ? mm : m;
  }
  if ((N & 1) && tid == 0) {
    long long b = batch[N - 1];
    m = b > m ? b : m;
  }

  block_accum(a0, a1, accp);

  int mi = (int)m;
  for (int o = 16; o > 0; o >>= 1) {
    int other = __shfl_xor(mi, o, 32);
    mi = other > mi ? other : mi;
  }
  __shared__ int mlds[8];
  const int wave = threadIdx.x >> 5;
  const int lane = threadIdx.x & 31;
  if (lane == 0) mlds[wave] = mi;
  __syncthreads();
  if (threadIdx.x == 0) {
    int t = mlds[0];
    const int nw = blockDim.x >> 5;
    for (int i = 1; i < nw; ++i) t = mlds[i] > t ? mlds[i] : t;
    atomicMax(bmax, t);
  }
}

__global__ void erdos_finalize_kernel(const double* __restrict__ acc,
                                      const int* __restrict__ bmax,
                                      const float* __restrict__ penalty,
                                      float* __restrict__ out) {
  if (threadIdx.x == 0 && blockIdx.x == 0) {
    const double ng = (double)(bmax[0] + 1);
    const double ed = acc[0] / ng;  // expected_distance
    const double ew = acc[1] / ng;  // expected_weight (segment_sum mean)
    const double loss = (double)penalty[0] * ed + ew;
    out[0] = (float)loss;
    out[1] = (float)ew;
    out[2] = (float)ed;
  }
}

extern "C" void kernel_launch(void* const* d_in, const int* in_sizes, int n_in,
                              void* d_out, int out_size, void* d_ws, size_t ws_size,
                              hipStream_t stream) {
  const float* probs = (const float*)d_in[0];
  const long long* edge_index = (const long long*)d_in[1];  // int64 [2, E]
  const long long* batch = (const long long*)d_in[2];       // int64 [N]
  const float* penalty = (const float*)d_in[3];             // scalar

  const long long N = (long long)in_sizes[0];
  const long long E = (long long)in_sizes[1] / 2;

  double* acc = (double*)d_ws;                         // [0]=edge sum, [1]=prob sum
  int* bmax = (int*)((char*)d_ws + 2 * sizeof(double));
  float* out = (float*)d_out;

  erdos_init_kernel<<<1, 1, 0, stream>>>(acc, bmax);

  const int threads = 256;  // 8 wave32 per block
  // ~1M threads for the 512MB edge stream: 16 b128-pair iters per thread.
  erdos_edge_kernel<<<4096, threads, 0, stream>>>(
      probs, edge_index, edge_index + E, E, acc);
  erdos_node_kernel<<<1024, threads, 0, stream>>>(probs, batch, N, acc + 1, bmax);
  erdos_finalize_kernel<<<1, 1, 0, stream>>>(acc, bmax, penalty, out);
}